// GCNImproved_14748917694873
// MI455X (gfx1250) — compile-verified
//
#include <hip/hip_runtime.h>
#include <hip/hip_bf16.h>

typedef __attribute__((ext_vector_type(2))) float v2f;
typedef __attribute__((ext_vector_type(8))) float v8f;

// ---------------------------------------------------------------------------
// Degree / normalization precompute
// ---------------------------------------------------------------------------
__global__ void deg_init_kernel(float* __restrict__ deg, int n) {
    int i = blockIdx.x * blockDim.x + threadIdx.x;
    if (i < n) deg[i] = 1.0f;  // self-loop contribution (ew = 1)
}

__global__ void deg_accum_kernel(const int* __restrict__ dst,
                                 const float* __restrict__ ew,
                                 float* __restrict__ deg, int e) {
    int i = blockIdx.x * blockDim.x + threadIdx.x;
    if (i < e) atomicAdd(&deg[dst[i]], ew[i]);
}

__global__ void dinv_kernel(float* __restrict__ deg, int n) {
    int i = blockIdx.x * blockDim.x + threadIdx.x;
    if (i < n) {
        float d = deg[i];
        deg[i] = (d > 0.0f) ? rsqrtf(d) : 0.0f;  // in-place deg -> dinv
    }
}

__global__ void norm_kernel(const int* __restrict__ src, const int* __restrict__ dst,
                            const float* __restrict__ ew, const float* __restrict__ dinv,
                            float* __restrict__ norm, int e) {
    int i = blockIdx.x * blockDim.x + threadIdx.x;
    if (i < e) norm[i] = dinv[src[i]] * ew[i] * dinv[dst[i]];
}

// ---------------------------------------------------------------------------
// Dense GEMM  C[N x 64] = A[N x 64] @ W[64 x 64] (+ bias)  via f32 WMMA
// Block = 256 threads = 8 waves; wave w handles rows [16*(8*blk+w), +16).
// Per wave: 4 accumulators (16x16 tiles over the 64 output cols), K in steps
// of 4 -> V_WMMA_F32_16X16X4_F32.
// ---------------------------------------------------------------------------
__global__ void gemm_wmma_kernel(const float* __restrict__ A,
                                 const float* __restrict__ W,
                                 const float* __restrict__ bias,
                                 float* __restrict__ C,
                                 int nrows, int has_bias) {
    __shared__ float sW[64 * 64];
    int tid = threadIdx.x;
    for (int i = tid; i < 64 * 64; i += 256) sW[i] = W[i];
    __syncthreads();

    int wave = tid >> 5;           // 0..7  (wave32)
    int lane = tid & 31;
    int rowTile = blockIdx.x * 8 + wave;
    int row0 = rowTile * 16;
    if (row0 >= nrows) return;     // wave-uniform exit: EXEC all-ones for WMMA

    int half = lane >> 4;          // 0: lanes 0-15, 1: lanes 16-31
    int l15  = lane & 15;

    // A fragment row for this lane (16x4 f32 layout: lanes 0-15 rows M=0..15
    // holding K pair {0,1}; lanes 16-31 same rows holding K pair {2,3})
    int arow = row0 + l15;
    if (arow >= nrows) arow = nrows - 1;  // per-lane clamp (select, no branch)
    const float* Arow = A + (size_t)arow * 64;

    v8f acc0 = {}, acc1 = {}, acc2 = {}, acc3 = {};

#pragma unroll
    for (int k = 0; k < 64; k += 4) {
        // A fragment: VGPR0 = K(k + 2*half), VGPR1 = K(k + 2*half + 1)
        v2f a = *(const v2f*)(Arow + k + half * 2);
        // B fragments per 16-col tile: row K striped over lanes (N = l15),
        // VGPR0 = K(k + 2*half), VGPR1 = K(k + 2*half + 1)
        v2f b0, b1, b2, b3;
        b0.x = sW[(k + half * 2 + 0) * 64 +  0 + l15];
        b0.y = sW[(k + half * 2 + 1) * 64 +  0 + l15];
        b1.x = sW[(k + half * 2 + 0) * 64 + 16 + l15];
        b1.y = sW[(k + half * 2 + 1) * 64 + 16 + l15];
        b2.x = sW[(k + half * 2 + 0) * 64 + 32 + l15];
        b2.y = sW[(k + half * 2 + 1) * 64 + 32 + l15];
        b3.x = sW[(k + half * 2 + 0) * 64 + 48 + l15];
        b3.y = sW[(k + half * 2 + 1) * 64 + 48 + l15];

        acc0 = __builtin_amdgcn_wmma_f32_16x16x4_f32(false, a, false, b0, (short)0, acc0, false, false);
        acc1 = __builtin_amdgcn_wmma_f32_16x16x4_f32(false, a, false, b1, (short)0, acc1, false, false);
        acc2 = __builtin_amdgcn_wmma_f32_16x16x4_f32(false, a, false, b2, (short)0, acc2, false, false);
        acc3 = __builtin_amdgcn_wmma_f32_16x16x4_f32(false, a, false, b3, (short)0, acc3, false, false);
    }

    float bv0 = 0.f, bv1 = 0.f, bv2 = 0.f, bv3 = 0.f;
    if (has_bias) {
        bv0 = bias[ 0 + l15];
        bv1 = bias[16 + l15];
        bv2 = bias[32 + l15];
        bv3 = bias[48 + l15];
    }

    // C/D layout: VGPR r holds M = 8*half + r, N = l15 (per 16-col tile)
    int srow = row0 + half * 8;
#pragma unroll
    for (int r = 0; r < 8; ++r) {
        int row = srow + r;
        if (row < nrows) {
            float* Crow = C + (size_t)row * 64;
            Crow[ 0 + l15] = acc0[r] + bv0;
            Crow[16 + l15] = acc1[r] + bv1;
            Crow[32 + l15] = acc2[r] + bv2;
            Crow[48 + l15] = acc3[r] + bv3;
        }
    }
}

// ---------------------------------------------------------------------------
// Scatter init: out[i,:] = dinv[i]^2 * hw[i,:] + b   (self-loop + bias)
// ---------------------------------------------------------------------------
__global__ void scatter_init_kernel(const float* __restrict__ hw,
                                    const float* __restrict__ dinv,
                                    const float* __restrict__ bias,
                                    float* __restrict__ out, int n) {
    int gid = blockIdx.x * blockDim.x + threadIdx.x;
    if (gid < n * 64) {
        int i = gid >> 6;
        int d = gid & 63;
        float di = dinv[i];
        out[gid] = di * di * hw[gid] + bias[d];
    }
}

// ---------------------------------------------------------------------------
// Edge scatter: out[dst,:] += norm[e] * hw[src,:]   (16 threads / edge)
// ---------------------------------------------------------------------------
__global__ void scatter_kernel(const float* __restrict__ hw,
                               const int* __restrict__ src,
                               const int* __restrict__ dst,
                               const float* __restrict__ norm,
                               float* __restrict__ out, int e) {
    int gid = blockIdx.x * blockDim.x + threadIdx.x;
    int edge = gid >> 4;
    if (edge >= e) return;
    int p = gid & 15;
    int s = src[edge];
    int d = dst[edge];
    float w = norm[edge];
    const float4 v = *(const float4*)(hw + (size_t)s * 64 + p * 4);
    float* o = out + (size_t)d * 64 + p * 4;
    atomicAdd(o + 0, w * v.x);
    atomicAdd(o + 1, w * v.y);
    atomicAdd(o + 2, w * v.z);
    atomicAdd(o + 3, w * v.w);
}

// ---------------------------------------------------------------------------
// LayerNorm + ReLU + residual: one wave32 per row, 2 elems per lane.
// ---------------------------------------------------------------------------
__global__ void ln_relu_res_kernel(const float* __restrict__ h,
                                   const float* __restrict__ res,
                                   const float* __restrict__ g,
                                   const float* __restrict__ be,
                                   float* __restrict__ out, int n) {
    int gid  = blockIdx.x * blockDim.x + threadIdx.x;
    int row  = gid >> 5;
    int lane = threadIdx.x & 31;
    if (row >= n) return;  // wave-uniform

    const float2 v = *(const float2*)(h + (size_t)row * 64 + lane * 2);
    float s = v.x + v.y;
#pragma unroll
    for (int off = 16; off > 0; off >>= 1) s += __shfl_xor(s, off, 32);
    float m = s * (1.0f / 64.0f);

    float dx = v.x - m, dy = v.y - m;
    float q = dx * dx + dy * dy;
#pragma unroll
    for (int off = 16; off > 0; off >>= 1) q += __shfl_xor(q, off, 32);
    float rinv = rsqrtf(q * (1.0f / 64.0f) + 1e-5f);

    const float2 r  = *(const float2*)(res + (size_t)row * 64 + lane * 2);
    const float2 gg = *(const float2*)(g + lane * 2);
    const float2 bb = *(const float2*)(be + lane * 2);
    float2 y;
    y.x = fmaxf(dx * rinv * gg.x + bb.x, 0.0f) + r.x;
    y.y = fmaxf(dy * rinv * gg.y + bb.y, 0.0f) + r.y;
    *(float2*)(out + (size_t)row * 64 + lane * 2) = y;
}

// ---------------------------------------------------------------------------
// Scores: one wave32 per (user,item) pair -> 64-dim dot + biases + clip
// ---------------------------------------------------------------------------
__global__ void score_kernel(const int* __restrict__ users,
                             const int* __restrict__ items,
                             const float* __restrict__ h,
                             const float* __restrict__ bu,
                             const float* __restrict__ bi,
                             const float* __restrict__ mu,
                             float* __restrict__ out, int nb, int nusers) {
    int gid  = blockIdx.x * blockDim.x + threadIdx.x;
    int b    = gid >> 5;
    int lane = threadIdx.x & 31;
    if (b >= nb) return;  // wave-uniform
    int u  = users[b];
    int it = items[b];
    const float2 hu = *(const float2*)(h + (size_t)u * 64 + lane * 2);
    const float2 hi = *(const float2*)(h + ((size_t)nusers + it) * 64 + lane * 2);
    float s = hu.x * hi.x + hu.y * hi.y;
#pragma unroll
    for (int off = 16; off > 0; off >>= 1) s += __shfl_xor(s, off, 32);
    if (lane == 0) {
        float r = s + bu[u] + bi[it] + mu[0];
        out[b] = fminf(fmaxf(r, 1.0f), 5.0f);
    }
}

// ---------------------------------------------------------------------------
extern "C" void kernel_launch(void* const* d_in, const int* in_sizes, int n_in,
                              void* d_out, int out_size, void* d_ws, size_t ws_size,
                              hipStream_t stream) {
    const int*   users = (const int*)d_in[0];
    const int*   items = (const int*)d_in[1];
    const int*   eidx  = (const int*)d_in[2];
    const float* ew    = (const float*)d_in[3];
    const float* U     = (const float*)d_in[4];
    const float* I     = (const float*)d_in[5];
    const float* W0    = (const float*)d_in[6];
    const float* b0    = (const float*)d_in[7];
    const float* g0    = (const float*)d_in[8];
    const float* be0   = (const float*)d_in[9];
    const float* W1    = (const float*)d_in[10];
    const float* b1    = (const float*)d_in[11];
    const float* g1    = (const float*)d_in[12];
    const float* be1   = (const float*)d_in[13];
    const float* Wp    = (const float*)d_in[14];
    const float* bp    = (const float*)d_in[15];
    const float* bu    = (const float*)d_in[16];
    const float* bi    = (const float*)d_in[17];
    const float* mu    = (const float*)d_in[18];
    float* out = (float*)d_out;

    const int E  = in_sizes[3];
    const int NU = in_sizes[16];
    const int NI = in_sizes[17];
    const int N  = NU + NI;
    const int B  = in_sizes[0];

    const int* src = eidx;
    const int* dst = eidx + E;

    // Workspace layout (floats): dinv[N] | norm[E] | buf0[N*64] | buf1 | buf2
    float* ws   = (float*)d_ws;
    float* dinv = ws;
    float* nrm  = ws + N;
    float* buf0 = nrm + E;
    float* buf1 = buf0 + (size_t)N * 64;
    float* buf2 = buf1 + (size_t)N * 64;

    // x = concat(U, I) -> buf0
    hipMemcpyAsync(buf0, U, (size_t)NU * 64 * sizeof(float), hipMemcpyDeviceToDevice, stream);
    hipMemcpyAsync(buf0 + (size_t)NU * 64, I, (size_t)NI * 64 * sizeof(float), hipMemcpyDeviceToDevice, stream);

    // Degrees -> dinv, edge norms (shared by both conv layers)
    deg_init_kernel <<<(N + 255) / 256, 256, 0, stream>>>(dinv, N);
    deg_accum_kernel<<<(E + 255) / 256, 256, 0, stream>>>(dst, ew, dinv, E);
    dinv_kernel     <<<(N + 255) / 256, 256, 0, stream>>>(dinv, N);
    norm_kernel     <<<(E + 255) / 256, 256, 0, stream>>>(src, dst, ew, dinv, nrm, E);

    const int gemm_blocks    = ((N + 15) / 16 + 7) / 8;
    const int init_blocks    = (N * 64 + 255) / 256;
    const int scatter_blocks = (E * 16 + 255) / 256;
    const int ln_blocks      = (N * 32 + 255) / 256;

    // Layer 1: conv(x, W0, b0) -> LN/ReLU + residual(x)
    gemm_wmma_kernel   <<<gemm_blocks, 256, 0, stream>>>(buf0, W0, nullptr, buf1, N, 0);
    scatter_init_kernel<<<init_blocks, 256, 0, stream>>>(buf1, dinv, b0, buf2, N);
    scatter_kernel     <<<scatter_blocks, 256, 0, stream>>>(buf1, src, dst, nrm, buf2, E);
    ln_relu_res_kernel <<<ln_blocks, 256, 0, stream>>>(buf2, buf0, g0, be0, buf1, N);   // h1 -> buf1

    // Layer 2: conv(h1, W1, b1) -> LN/ReLU + residual(h1)
    gemm_wmma_kernel   <<<gemm_blocks, 256, 0, stream>>>(buf1, W1, nullptr, buf2, N, 0);
    scatter_init_kernel<<<init_blocks, 256, 0, stream>>>(buf2, dinv, b1, buf0, N);
    scatter_kernel     <<<scatter_blocks, 256, 0, stream>>>(buf2, src, dst, nrm, buf0, E);
    ln_relu_res_kernel <<<ln_blocks, 256, 0, stream>>>(buf0, buf1, g1, be1, buf2, N);   // h2 -> buf2

    // Projection: h3 = h2 @ Wp + bp -> buf0
    gemm_wmma_kernel<<<gemm_blocks, 256, 0, stream>>>(buf2, Wp, bp, buf0, N, 1);

    // Scores
    score_kernel<<<(B * 32 + 255) / 256, 256, 0, stream>>>(users, items, buf0, bu, bi, mu, out, B, NU);
}